// FiLMEncoder_56538949485251
// MI455X (gfx1250) — compile-verified
//
#include <hip/hip_runtime.h>
#include <hip/hip_bf16.h>

#define NODES   100000
#define EDGES   1600000
#define DCH     128
#define NLAYERS 4

// Packed-B geometry: 48 col-tiles x 32 k-groups x 32 lanes x float2
#define NTILES     48
#define KGROUPS    32
#define BP_PER_L   (NTILES * KGROUPS * 32 * 2)   // 98304 floats per layer

typedef __attribute__((ext_vector_type(2))) float v2f;
typedef __attribute__((ext_vector_type(8))) float v8f;

// ---------------------------------------------------------------------------
// Utility kernels
// ---------------------------------------------------------------------------
__global__ void zero_f32(float* __restrict__ p, int n) {
    int i = blockIdx.x * blockDim.x + threadIdx.x;
    if (i < n) p[i] = 0.0f;
}

__global__ void degree_kernel(const int* __restrict__ ei, float* __restrict__ deg) {
    int e = blockIdx.x * blockDim.x + threadIdx.x;
    if (e < EDGES) {
        int dst = ei[EDGES + e];
        atomicAdd(&deg[dst], 1.0f);
    }
}

__global__ void deginv_kernel(float* __restrict__ deg) {
    int i = blockIdx.x * blockDim.x + threadIdx.x;
    if (i < NODES) deg[i] = 1.0f / fmaxf(deg[i], 1.0f);
}

__global__ void act_kernel(const float* __restrict__ in, float* __restrict__ out,
                           int n, int do_relu) {
    int i = blockIdx.x * blockDim.x + threadIdx.x;
    if (i < n) {
        float v = in[i];
        out[i] = do_relu ? fmaxf(v, 0.0f) : v;
    }
}

// ---------------------------------------------------------------------------
// Pack all layer weights into the WMMA-native B layout:
//   Bp[l][ct][g][lane] = { W[4g+koff][j], W[4g+koff+1][j] }
// with col = lane&15, koff = (lane>>4)*2, j = ct*16+col mapped onto
// [W_lin(128) | W_film(256) | W_skip(128) | W_fskip(256)].
// Run once per call; 1.5 MB total, stays hot in L2 for the whole GEMM phase.
// ---------------------------------------------------------------------------
__global__ void pack_weights(const float* __restrict__ W_lin,
                             const float* __restrict__ W_film,
                             const float* __restrict__ W_skip,
                             const float* __restrict__ W_fskp,
                             float* __restrict__ Bp)
{
    const int l   = blockIdx.y;
    const int idx = blockIdx.x * blockDim.x + threadIdx.x;    // per-layer 48*32*32
    if (idx >= NTILES * KGROUPS * 32) return;

    const int lane = idx & 31;
    const int g    = (idx >> 5) & 31;
    const int ct   = idx >> 10;

    const int col  = lane & 15;
    const int koff = (lane >> 4) * 2;
    const int k    = 4 * g + koff;
    const int j    = ct * 16 + col;

    const float* Wl  = W_lin  + (size_t)l * 128 * 128;
    const float* Wf  = W_film + (size_t)l * 128 * 256;
    const float* Ws  = W_skip + (size_t)l * 128 * 128;
    const float* Wfs = W_fskp + (size_t)l * 128 * 256;

    const float* Wp; int jj; int stride;
    if (j < 128)      { Wp = Wl;  jj = j;       stride = 128; }
    else if (j < 384) { Wp = Wf;  jj = j - 128; stride = 256; }
    else if (j < 512) { Wp = Ws;  jj = j - 384; stride = 128; }
    else              { Wp = Wfs; jj = j - 512; stride = 256; }

    float* out = Bp + (size_t)l * BP_PER_L + (size_t)idx * 2;
    out[0] = Wp[k * stride + jj];
    out[1] = Wp[(k + 1) * stride + jj];
}

// ---------------------------------------------------------------------------
// Strip GEMM: each block handles 16 nodes, producing all 768 columns of
//   Y = X * [W_lin | W_film | W_skip | W_fskip]
// Steady state per WMMA step: 1x ds_load_b64 (A) + 1x global_load_b64 (B,
// immediate offset, coalesced) + 1x v_wmma_f32_16x16x4_f32.
// ---------------------------------------------------------------------------
__global__ __launch_bounds__(256) void film_gemm_strip(
    const float* __restrict__ x,     // NODES x 128
    const float* __restrict__ Bp,    // packed weights for this layer
    const float* __restrict__ bf,    // 256 bias
    float* __restrict__ h_g,
    float* __restrict__ gamma_g,
    float* __restrict__ beta_g,
    float* __restrict__ out_g)
{
    __shared__ float ldsA[16 * 132];   // padded stride: conflict-free WMMA A reads
    __shared__ float ldsY[16 * 768];   // full-width result strip

    const int tid      = threadIdx.x;
    const int wave     = tid >> 5;
    const int lane     = tid & 31;
    const int nodeBase = blockIdx.x * 16;

    // Stage 16x128 X strip into LDS (float2 granularity, 8B-aligned)
    for (int i = tid; i < 16 * 64; i += 256) {
        int r = i >> 6, c2 = i & 63;
        const float2 v = *(const float2*)&x[(nodeBase + r) * DCH + c2 * 2];
        *(float2*)&ldsA[r * 132 + c2 * 2] = v;
    }
    __syncthreads();

    // A-matrix fp32 16x4 layout: lanes 0-15 -> K pair (0,1); lanes 16-31 -> (2,3)
    const int row  = lane & 15;
    const int koff = (lane >> 4) * 2;
    const float* At = &ldsA[row * 132 + koff];

    for (int t = 0; t < 6; ++t) {
        const int ct = wave + 8 * t;              // column tile 0..47
        const int j0 = ct * 16;
        const float* Bt = Bp + ct * (KGROUPS * 64) + lane * 2;

        v8f acc = {};
        #pragma unroll
        for (int g = 0; g < KGROUPS; ++g) {
            v2f a = *(const v2f*)(At + 4 * g);     // ds_load_b64, imm offset 16*g
            v2f b = *(const v2f*)(Bt + 64 * g);    // global_load_b64, imm offset 256*g
            acc = __builtin_amdgcn_wmma_f32_16x16x4_f32(
                false, a, false, b, (short)0, acc, false, false);
        }

        // D layout: VGPR r -> row r (lanes 0-15) / row r+8 (lanes 16-31), col = lane&15
        const int mbase = (lane >> 4) * 8;
        const int n     = lane & 15;
        #pragma unroll
        for (int r = 0; r < 8; ++r) {
            ldsY[(mbase + r) * 768 + j0 + n] = acc[r];
        }
    }
    __syncthreads();

    // FiLM epilogue over the 16x768 strip
    for (int i = tid; i < 16 * 128; i += 256) {
        int r = i >> 7, c = i & 127;
        const float* yr = &ldsY[r * 768];
        int node = nodeBase + r;

        float h     = yr[c];
        float beta  = yr[128 + c] + bf[c];
        float gamma = yr[256 + c] + bf[128 + c];
        float ys    = yr[384 + c];
        float bs    = yr[512 + c];
        float gs    = yr[640 + c];
        float o     = fmaxf(gs * ys + bs, 0.0f);

        h_g[node * DCH + c]     = h;
        beta_g[node * DCH + c]  = beta;
        gamma_g[node * DCH + c] = gamma;
        out_g[node * DCH + c]   = o;   // skip term; edge kernel accumulates on top
    }
}

// ---------------------------------------------------------------------------
// Edge phase: 32 lanes per edge, 4 channels per lane.
//   msg = relu(gamma[dst]*h[src] + beta[dst]) * deg_inv[dst]; out[dst] += msg
// h/gamma/beta (3 x 51 MB) stay L2-resident on MI455X (192 MB L2).
// ---------------------------------------------------------------------------
__global__ __launch_bounds__(256) void film_edge(
    const int* __restrict__ ei,
    const float* __restrict__ h,
    const float* __restrict__ gamma,
    const float* __restrict__ beta,
    const float* __restrict__ deg_inv,
    float* __restrict__ out)
{
    int idx = blockIdx.x * blockDim.x + threadIdx.x;   // EDGES*32 = 51.2M < 2^31
    int e = idx >> 5;
    if (e >= EDGES) return;
    int c = (idx & 31) * 4;

    int src = ei[e];
    int dst = ei[EDGES + e];

    const float4 hv = *(const float4*)&h[src * DCH + c];
    const float4 gv = *(const float4*)&gamma[dst * DCH + c];
    const float4 bv = *(const float4*)&beta[dst * DCH + c];
    float di = deg_inv[dst];

    float m0 = fmaxf(gv.x * hv.x + bv.x, 0.0f) * di;
    float m1 = fmaxf(gv.y * hv.y + bv.y, 0.0f) * di;
    float m2 = fmaxf(gv.z * hv.z + bv.z, 0.0f) * di;
    float m3 = fmaxf(gv.w * hv.w + bv.w, 0.0f) * di;

    float* op = &out[dst * DCH + c];
    atomicAdd(op + 0, m0);
    atomicAdd(op + 1, m1);
    atomicAdd(op + 2, m2);
    atomicAdd(op + 3, m3);
}

// ---------------------------------------------------------------------------
// Host-side launch
// ---------------------------------------------------------------------------
extern "C" void kernel_launch(void* const* d_in, const int* in_sizes, int n_in,
                              void* d_out, int out_size, void* d_ws, size_t ws_size,
                              hipStream_t stream) {
    const float* x0     = (const float*)d_in[0];
    const int*   ei     = (const int*)d_in[1];
    const float* W_lin  = (const float*)d_in[2];   // 4 x 128 x 128
    const float* W_film = (const float*)d_in[3];   // 4 x 128 x 256
    const float* b_film = (const float*)d_in[4];   // 4 x 256
    const float* W_skip = (const float*)d_in[5];   // 4 x 128 x 128
    const float* W_fskp = (const float*)d_in[6];   // 4 x 128 x 256

    float* w = (float*)d_ws;
    float* deg   = w; w += NODES;                  // degree -> deg_inv (in place)
    float* xA    = w; w += (size_t)NODES * DCH;
    float* hB    = w; w += (size_t)NODES * DCH;
    float* gamB  = w; w += (size_t)NODES * DCH;
    float* betB  = w; w += (size_t)NODES * DCH;
    float* outB  = w; w += (size_t)NODES * DCH;
    float* BpAll = w; w += (size_t)NLAYERS * BP_PER_L;

    // Pack weights into WMMA-native B layout (all layers)
    {
        dim3 grid((NTILES * KGROUPS * 32 + 255) / 256, NLAYERS);
        pack_weights<<<grid, 256, 0, stream>>>(W_lin, W_film, W_skip, W_fskp, BpAll);
    }

    // Degrees (recomputed every call: deterministic, no state carried)
    zero_f32<<<(NODES + 255) / 256, 256, 0, stream>>>(deg, NODES);
    degree_kernel<<<(EDGES + 255) / 256, 256, 0, stream>>>(ei, deg);
    deginv_kernel<<<(NODES + 255) / 256, 256, 0, stream>>>(deg);

    const int gemmBlocks = NODES / 16;                    // 6250 (exact)
    const int edgeBlocks = (EDGES * 32) / 256;            // 200000 (exact)
    const int nElem      = NODES * DCH;

    const float* xin = x0;
    for (int l = 0; l < NLAYERS; ++l) {
        const float* Bp = BpAll + (size_t)l * BP_PER_L;
        const float* bf = b_film + (size_t)l * 256;

        film_gemm_strip<<<gemmBlocks, 256, 0, stream>>>(
            xin, Bp, bf, hB, gamB, betB, outB);

        film_edge<<<edgeBlocks, 256, 0, stream>>>(ei, hB, gamB, betB, deg, outB);

        if (l < NLAYERS - 1) {
            act_kernel<<<(nElem + 255) / 256, 256, 0, stream>>>(outB, xA, nElem, 1);
            xin = xA;
        } else {
            act_kernel<<<(nElem + 255) / 256, 256, 0, stream>>>(outB, (float*)d_out, nElem, 0);
        }
    }
}